// GINLayer_16423954940358
// MI455X (gfx1250) — compile-verified
//
#include <hip/hip_runtime.h>

#define D 128
#define BN_EPS 1e-5f

typedef __attribute__((ext_vector_type(2))) float v2f;
typedef __attribute__((ext_vector_type(8))) float v8f;

// ---------------------------------------------------------------- utilities
__global__ void zero_kernel(float* __restrict__ p, int n) {
  int i = blockIdx.x * blockDim.x + threadIdx.x;
  if (i < n) p[i] = 0.0f;
}

// ------------------------------------------------- edge scatter-add (1 wave/edge)
__global__ void scatter_add_kernel(const float* __restrict__ hsrc,
                                   const int* __restrict__ src,
                                   const int* __restrict__ dst,
                                   float* __restrict__ x, int E) {
  long t = (long)blockIdx.x * blockDim.x + threadIdx.x;
  int e = (int)(t >> 5);
  if (e >= E) return;
  int lane = (int)(t & 31);
  int s = src[e];
  int d = dst[e];
  const float4 v = *reinterpret_cast<const float4*>(hsrc + (size_t)s * D + lane * 4);
  float* o = x + (size_t)d * D + lane * 4;
  unsafeAtomicAdd(o + 0, v.x);
  unsafeAtomicAdd(o + 1, v.y);
  unsafeAtomicAdd(o + 2, v.z);
  unsafeAtomicAdd(o + 3, v.w);
}

// ------------------------------------------------- f32 WMMA GEMM  Out = preop(A) @ W
// A: [N,128] row-major, W: [128,128] row-major (k-major), Out: [N,128]
// 1 block = 16 rows of A staged in LDS; 8 waves -> 8 column tiles of 16.
// applyAffine: A' = relu(A*scale[k] + shift[k]) applied at LDS-fill time.
__global__ void __launch_bounds__(256)
gemm_wmma_f32(const float* __restrict__ A, const float* __restrict__ W,
              float* __restrict__ Out,
              const float* __restrict__ scale, const float* __restrict__ shift,
              int N, int applyAffine) {
  __shared__ float lA[16 * D];
  const int rowBase = blockIdx.x * 16;

  // ---- stage 16x128 strip of A into LDS (each thread: 8 floats)
  {
    int t = threadIdx.x;           // 0..255
    int r = t >> 4;                // 0..15 local row
    int k0 = (t & 15) * 8;         // 0..120
    int gr = rowBase + r;
    float4 v0 = {0.f, 0.f, 0.f, 0.f};
    float4 v1 = {0.f, 0.f, 0.f, 0.f};
    if (gr < N) {
      const float* p = A + (size_t)gr * D + k0;
      v0 = *reinterpret_cast<const float4*>(p);
      v1 = *reinterpret_cast<const float4*>(p + 4);
    }
    if (applyAffine) {
      float4 s0 = *reinterpret_cast<const float4*>(scale + k0);
      float4 s1 = *reinterpret_cast<const float4*>(scale + k0 + 4);
      float4 h0 = *reinterpret_cast<const float4*>(shift + k0);
      float4 h1 = *reinterpret_cast<const float4*>(shift + k0 + 4);
      v0.x = fmaxf(v0.x * s0.x + h0.x, 0.f);
      v0.y = fmaxf(v0.y * s0.y + h0.y, 0.f);
      v0.z = fmaxf(v0.z * s0.z + h0.z, 0.f);
      v0.w = fmaxf(v0.w * s0.w + h0.w, 0.f);
      v1.x = fmaxf(v1.x * s1.x + h1.x, 0.f);
      v1.y = fmaxf(v1.y * s1.y + h1.y, 0.f);
      v1.z = fmaxf(v1.z * s1.z + h1.z, 0.f);
      v1.w = fmaxf(v1.w * s1.w + h1.w, 0.f);
    }
    *reinterpret_cast<float4*>(&lA[r * D + k0]) = v0;
    *reinterpret_cast<float4*>(&lA[r * D + k0 + 4]) = v1;
  }
  __syncthreads();

  const int wave = threadIdx.x >> 5;
  const int lane = threadIdx.x & 31;
  const int colBase = wave * 16;
  const int col = colBase + (lane & 15);   // N index within W / Out
  const int kb = (lane >> 4) * 2;          // lanes 16-31 carry K=2,3 of each quad
  const int arow = lane & 15;              // A-matrix row for this lane

  v8f acc = {};
  const float* wp = W + col;

#pragma unroll
  for (int k = 0; k < D; k += 4) {
    // A 16x4 tile: VGPR0 = K=kb, VGPR1 = K=kb+1 (consecutive in LDS row)
    v2f a = *reinterpret_cast<const v2f*>(&lA[arow * D + k + kb]);
    // B 4x16 tile: per-lane column `col`, K rows kb / kb+1 of this quad
    v2f b;
    b.x = wp[(size_t)(k + kb) * D];
    b.y = wp[(size_t)(k + kb + 1) * D];
    acc = __builtin_amdgcn_wmma_f32_16x16x4_f32(
        /*neg_a=*/false, a, /*neg_b=*/false, b,
        /*c_mod=*/(short)0, acc, /*reuse_a=*/false, /*reuse_b=*/false);
  }

  // C/D layout: VGPR i, lanes 0-15 -> M=i ; lanes 16-31 -> M=i+8 ; N=lane&15
  const int mhi = (lane >> 4) * 8;
#pragma unroll
  for (int i = 0; i < 8; ++i) {
    int gr = rowBase + i + mhi;
    if (gr < N) Out[(size_t)gr * D + col] = acc[i];
  }
}

// ------------------------------------------------- per-column sum / sumsq
__global__ void __launch_bounds__(256)
col_stats_kernel(const float* __restrict__ Y, int N, float* __restrict__ sums) {
  __shared__ float ls[256];
  __shared__ float lq[256];
  int c = threadIdx.x & (D - 1);
  int g = threadIdx.x >> 7;  // 0..1
  float s = 0.f, q = 0.f;
  for (int r = blockIdx.x * 2 + g; r < N; r += gridDim.x * 2) {
    float v = Y[(size_t)r * D + c];
    s += v;
    q += v * v;
  }
  ls[threadIdx.x] = s;
  lq[threadIdx.x] = q;
  __syncthreads();
  if (threadIdx.x < D) {
    s = ls[threadIdx.x] + ls[threadIdx.x + D];
    q = lq[threadIdx.x] + lq[threadIdx.x + D];
    unsafeAtomicAdd(&sums[c], s);
    unsafeAtomicAdd(&sums[D + c], q);
  }
}

// ------------------------------------------------- fold BN stats into scale/shift
__global__ void finalize_stats_kernel(const float* __restrict__ sums,
                                      const float* __restrict__ g,
                                      const float* __restrict__ b,
                                      float* __restrict__ scale,
                                      float* __restrict__ shift, int N) {
  int c = threadIdx.x;
  if (c < D) {
    float invN = 1.0f / (float)N;
    float mean = sums[c] * invN;
    float var = fmaxf(sums[D + c] * invN - mean * mean, 0.0f);  // biased var
    float inv = rsqrtf(var + BN_EPS);
    float sc = g[c] * inv;
    scale[c] = sc;
    shift[c] = b[c] - mean * sc;
  }
}

// ------------------------------------------------- out = relu(t*scale + shift)
__global__ void epilogue_kernel(const float* __restrict__ T,
                                const float* __restrict__ scale,
                                const float* __restrict__ shift,
                                float* __restrict__ out, long total4) {
  long i = (long)blockIdx.x * blockDim.x + threadIdx.x;
  if (i >= total4) return;
  int c = ((int)(i & 31)) * 4;
  float4 v = reinterpret_cast<const float4*>(T)[i];
  float4 s = *reinterpret_cast<const float4*>(scale + c);
  float4 h = *reinterpret_cast<const float4*>(shift + c);
  float4 o;
  o.x = fmaxf(v.x * s.x + h.x, 0.f);
  o.y = fmaxf(v.y * s.y + h.y, 0.f);
  o.z = fmaxf(v.z * s.z + h.z, 0.f);
  o.w = fmaxf(v.w * s.w + h.w, 0.f);
  reinterpret_cast<float4*>(out)[i] = o;
}

// ------------------------------------------------- one GIN relation
static void run_relation(const float* h_src, const float* h_dst,
                         const int* src, const int* dst, int E, int Ndst,
                         const float* W1, const float* W2,
                         const float* g1, const float* b1,
                         const float* g2, const float* b2,
                         float* outp, float* xbuf, float* ybuf, float* stats,
                         hipStream_t stream) {
  // x = h_dst + segment_sum(h_src[src] -> dst)
  hipMemcpyAsync(xbuf, h_dst, (size_t)Ndst * D * sizeof(float),
                 hipMemcpyDeviceToDevice, stream);
  {
    long tthreads = (long)E * 32;
    int blocks = (int)((tthreads + 255) / 256);
    scatter_add_kernel<<<blocks, 256, 0, stream>>>(h_src, src, dst, xbuf, E);
  }
  int tiles = (Ndst + 15) / 16;
  float* scl = stats + 2 * D;
  float* shf = stats + 3 * D;

  // y = x @ W1
  gemm_wmma_f32<<<tiles, 256, 0, stream>>>(xbuf, W1, ybuf, nullptr, nullptr, Ndst, 0);
  // BN1 stats
  zero_kernel<<<1, 256, 0, stream>>>(stats, 2 * D);
  col_stats_kernel<<<256, 256, 0, stream>>>(ybuf, Ndst, stats);
  finalize_stats_kernel<<<1, D, 0, stream>>>(stats, g1, b1, scl, shf, Ndst);
  // t = relu(BN1(y)) @ W2   (affine+relu fused into A load; t overwrites xbuf)
  gemm_wmma_f32<<<tiles, 256, 0, stream>>>(ybuf, W2, xbuf, scl, shf, Ndst, 1);
  // BN2 stats
  zero_kernel<<<1, 256, 0, stream>>>(stats, 2 * D);
  col_stats_kernel<<<256, 256, 0, stream>>>(xbuf, Ndst, stats);
  finalize_stats_kernel<<<1, D, 0, stream>>>(stats, g2, b2, scl, shf, Ndst);
  // out = relu(BN2(t))
  long total4 = (long)Ndst * (D / 4);
  int eb = (int)((total4 + 255) / 256);
  epilogue_kernel<<<eb, 256, 0, stream>>>(xbuf, scl, shf, outp, total4);
}

extern "C" void kernel_launch(void* const* d_in, const int* in_sizes, int n_in,
                              void* d_out, int out_size, void* d_ws, size_t ws_size,
                              hipStream_t stream) {
  const float* h_user   = (const float*)d_in[0];
  const float* h_item   = (const float*)d_in[1];
  const int* src_rates  = (const int*)d_in[2];
  const int* dst_rates  = (const int*)d_in[3];
  const int* src_rev    = (const int*)d_in[4];
  const int* dst_rev    = (const int*)d_in[5];
  const float* W1_rates = (const float*)d_in[6];
  const float* W2_rates = (const float*)d_in[7];
  const float* g1_rates = (const float*)d_in[8];
  const float* b1_rates = (const float*)d_in[9];
  const float* g2_rates = (const float*)d_in[10];
  const float* b2_rates = (const float*)d_in[11];
  const float* W1_rev   = (const float*)d_in[12];
  const float* W2_rev   = (const float*)d_in[13];
  const float* g1_rev   = (const float*)d_in[14];
  const float* b1_rev   = (const float*)d_in[15];
  const float* g2_rev   = (const float*)d_in[16];
  const float* b2_rev   = (const float*)d_in[17];

  int NU = in_sizes[0] / D;
  int NI = in_sizes[1] / D;
  int E  = in_sizes[2];

  float* out_user = (float*)d_out;
  float* out_item = out_user + (size_t)NU * D;

  int maxN = NU > NI ? NU : NI;
  float* xbuf  = (float*)d_ws;
  float* ybuf  = xbuf + (size_t)maxN * D;
  float* stats = ybuf + (size_t)maxN * D;  // 2*D sums + D scale + D shift

  // relation 'rates': user -> item  => out_item
  run_relation(h_user, h_item, src_rates, dst_rates, E, NI,
               W1_rates, W2_rates, g1_rates, b1_rates, g2_rates, b2_rates,
               out_item, xbuf, ybuf, stats, stream);
  // relation 'rated_by': item -> user => out_user
  run_relation(h_item, h_user, src_rev, dst_rev, E, NU,
               W1_rev, W2_rev, g1_rev, b1_rev, g2_rev, b2_rev,
               out_user, xbuf, ybuf, stats, stream);
}